// CombinedAttention_63161789055238
// MI455X (gfx1250) — compile-verified
//
#include <hip/hip_runtime.h>

#define BS 4
#define KK 10
#define LL 256
#define TT 32
#define DD 256
#define KL (KK * LL)   // 2560

typedef __attribute__((ext_vector_type(2))) float v2f;
typedef __attribute__((ext_vector_type(8))) float v8f;

// ---------------------------------------------------------------------------
// gfx1250 has native v_tanh_f32 (TRANS pipe, co-executes with VALU).
// Fallback: tanh(x) = 1 - 2/(exp2(x*2*log2 e) + 1) via v_exp_f32.
// ---------------------------------------------------------------------------
__device__ __forceinline__ float fast_exp2(float x) {
#if __has_builtin(__builtin_amdgcn_exp2f)
  return __builtin_amdgcn_exp2f(x);
#else
  return exp2f(x);
#endif
}

__device__ __forceinline__ float fast_tanh(float x) {
#if __has_builtin(__builtin_amdgcn_tanhf)
  return __builtin_amdgcn_tanhf(x);
#else
  float e = fast_exp2(2.885390081777927f * x);  // exp(2x)
  return 1.0f - 2.0f / (e + 1.0f);
#endif
}

// ---------------------------------------------------------------------------
// Async global->LDS DMA (VGLOBAL GLOBAL_LOAD_ASYNC_TO_LDS_B128, ASYNCcnt).
// Each lane copies 16B from its global address to its LDS byte offset.
// Generic-pointer low 32 bits == LDS offset (ISA 10.2 aperture rules).
// ---------------------------------------------------------------------------
__device__ __forceinline__ void async_copy_b128(unsigned lds_off,
                                                const float* gptr) {
  asm volatile("global_load_async_to_lds_b128 %0, %1, off"
               :: "v"(lds_off), "v"(gptr)
               : "memory");
}

__device__ __forceinline__ void wait_async_le2() {
#if __has_builtin(__builtin_amdgcn_s_wait_asynccnt)
  __builtin_amdgcn_s_wait_asynccnt(2);
#else
  asm volatile("s_wait_asynccnt 0x2" ::: "memory");
#endif
}

__device__ __forceinline__ void wait_async_le0() {
#if __has_builtin(__builtin_amdgcn_s_wait_asynccnt)
  __builtin_amdgcn_s_wait_asynccnt(0);
#else
  asm volatile("s_wait_asynccnt 0x0" ::: "memory");
#endif
}

// ---------------------------------------------------------------------------
// Kernel 1: qproj[b,t,e] = sum_d q[b,t,d]*Wq[e,d] + bq[e]   (tiny)
// ---------------------------------------------------------------------------
__global__ void qproj_kernel(const float* __restrict__ q,
                             const float* __restrict__ Wq,
                             const float* __restrict__ bq,
                             float* __restrict__ qproj) {
  __shared__ float qrow[DD];
  const int bt = blockIdx.x;       // (b*TT + t)
  const int e  = threadIdx.x;      // 0..255
  qrow[e] = q[(size_t)bt * DD + e];
  __syncthreads();
  const float* w = Wq + (size_t)e * DD;
  float acc = 0.0f;
#pragma unroll 8
  for (int d = 0; d < DD; ++d) acc += qrow[d] * w[d];
  qproj[(size_t)bt * DD + e] = acc + bq[e];
}

// ---------------------------------------------------------------------------
// Kernel 2: kproj[b] (KLxD) = kf[b] (KLxD) @ Wk^T (DxD), f32 WMMA 16x16x4.
// One 16x16 output tile per wave; K loop of 64 WMMAs, contiguous b64 loads.
// ---------------------------------------------------------------------------
__global__ void kproj_wmma_kernel(const float* __restrict__ kf,
                                  const float* __restrict__ Wk,
                                  float* __restrict__ kproj) {
  const int lane = threadIdx.x & 31;
  const int wave = threadIdx.x >> 5;
  const int tile = blockIdx.x * 8 + wave;     // 0..10239
  const int b  = tile / (160 * 16);
  const int r  = tile % (160 * 16);
  const int mt = r >> 4;                      // 0..159
  const int nt = r & 15;                      // 0..15

  const float* A = kf + (size_t)b * KL * DD;
  const int m0 = mt * 16, n0 = nt * 16;
  const int lm   = lane & 15;
  const int koff = (lane >> 4) << 1;          // 0 (lo half) or 2 (hi half)
  const float* arow = A  + (size_t)(m0 + lm) * DD + koff;
  const float* brow = Wk + (size_t)(n0 + lm) * DD + koff;

  v8f acc = {};
  for (int k0 = 0; k0 < DD; k0 += 4) {
    v2f a  = *(const v2f*)(arow + k0);
    v2f bm = *(const v2f*)(brow + k0);
    acc = __builtin_amdgcn_wmma_f32_16x16x4_f32(
        /*neg_a=*/false, a, /*neg_b=*/false, bm,
        /*c_mod=*/(short)0, acc, /*reuse_a=*/false, /*reuse_b=*/false);
  }

  // C layout: VGPR rr -> M = rr (lanes 0-15) / rr+8 (lanes 16-31), N = lane&15
  float* C = kproj + (size_t)b * KL * DD;
  const int col   = n0 + lm;
  const int rbase = m0 + ((lane >> 4) << 3);
#pragma unroll
  for (int rr = 0; rr < 8; ++rr)
    C[(size_t)(rbase + rr) * DD + col] = acc[rr];
}

// ---------------------------------------------------------------------------
// Kernel 3: scores[b][l][t] = sum_d tanh(kproj[b,l,d]+qproj[b,t,d])*we[d]
// Dominant kernel: 84M tanh. lanes = t (wave32 == T). qproj[b] in padded LDS
// (stride 257 -> conflict-free). Each wave double-buffers its OWN kproj row
// via GLOBAL_LOAD_ASYNC_TO_LDS_B128 (per-wave, in-order ASYNCcnt), so the
// hot loop has zero block barriers: issue row rr+1, s_wait_asynccnt<=2
// (retires row rr), tanh-reduce out of LDS while the next DMA flies.
// grid = BS*(KL/64) = 160 blocks, blockDim = 256 (8 waves, 8 rows each)
// ---------------------------------------------------------------------------
__global__ void scores_kernel(const float* __restrict__ kproj,
                              const float* __restrict__ qproj,
                              const float* __restrict__ we,
                              const int* __restrict__ mask,
                              float* __restrict__ scores) {
  __shared__ float qp[TT][DD + 1];   // +1 pad: stride 257 words, no conflicts
  __shared__ float wes[DD];
  __shared__ float kp[8][2][DD];     // [wave][double-buffer][d]

  const int tid   = threadIdx.x;
  const int b     = blockIdx.x / (KL / 64);
  const int lbase = (blockIdx.x % (KL / 64)) * 64;
  const int wave  = tid >> 5;
  const int t     = tid & 31;                 // lane == query index

  const float* kpg = kproj + (size_t)b * KL * DD;

  // Prologue: each wave starts the DMA of its first row into buffer 0.
  {
    unsigned lo = (unsigned)(uintptr_t)&kp[wave][0][0];
    const float* g = kpg + (size_t)(lbase + wave) * DD;
    async_copy_b128(lo + (unsigned)(t * 16), g + t * 4);
    async_copy_b128(lo + 512u + (unsigned)(t * 16), g + 128 + t * 4);
  }

  // Stage qproj[b] (32x256) and we while the first DMA is in flight.
  for (int i = tid; i < TT * DD; i += 256) {
    int tt = i >> 8, d = i & 255;
    qp[tt][d] = qproj[((size_t)b * TT + tt) * DD + d];
  }
  wes[tid] = we[tid];
  __syncthreads();

  for (int rr = 0; rr < 8; ++rr) {
    if (rr < 7) {  // kick off next row into the other buffer
      unsigned lo = (unsigned)(uintptr_t)&kp[wave][(rr + 1) & 1][0];
      const float* g = kpg + (size_t)(lbase + (rr + 1) * 8 + wave) * DD;
      async_copy_b128(lo + (unsigned)(t * 16), g + t * 4);
      async_copy_b128(lo + 512u + (unsigned)(t * 16), g + 128 + t * 4);
      wait_async_le2();   // row rr (issued first, in-order) has landed in LDS
    } else {
      wait_async_le0();
    }

    const int l = lbase + rr * 8 + wave;
    const float* kr = kp[wave][rr & 1];
    float acc = 0.0f;
#pragma unroll 8
    for (int d = 0; d < DD; ++d)
      acc += fast_tanh(kr[d] + qp[t][d]) * wes[d];

    if (mask[b * KL + l] == 0) acc = -1e9f;
    scores[((size_t)b * KL + l) * TT + t] = acc;   // 32 contiguous floats/wave
  }
}

// ---------------------------------------------------------------------------
// Kernel 4: fused softmax + beta reweight + renorm.
// out[b,t,l] = e^{s-max} * beta[b, l/LL] / sum_l (e^{s-max} * beta)
// (the intermediate softmax normalizer cancels). One block per (b,t).
// ---------------------------------------------------------------------------
__global__ void softmax_kernel(const float* __restrict__ scores,
                               const float* __restrict__ betas,
                               float* __restrict__ malphas) {
  __shared__ float red[256];
  const int bt = blockIdx.x;
  const int b = bt / TT, t = bt % TT;
  const int tid = threadIdx.x;

  float sv[10];
  float mx = -3.4e38f;
#pragma unroll
  for (int j = 0; j < 10; ++j) {
    int l = tid + j * 256;
    sv[j] = scores[((size_t)b * KL + l) * TT + t];
    mx = fmaxf(mx, sv[j]);
  }
  red[tid] = mx; __syncthreads();
  for (int s = 128; s > 0; s >>= 1) {
    if (tid < s) red[tid] = fmaxf(red[tid], red[tid + s]);
    __syncthreads();
  }
  mx = red[0]; __syncthreads();

  float sum = 0.0f;
#pragma unroll
  for (int j = 0; j < 10; ++j) {
    int l = tid + j * 256;
    float w = fast_exp2(1.4426950408889634f * (sv[j] - mx));
    w *= betas[b * KK + (l >> 8)];            // l / LL, LL == 256
    sv[j] = w;
    sum += w;
  }
  red[tid] = sum; __syncthreads();
  for (int s = 128; s > 0; s >>= 1) {
    if (tid < s) red[tid] += red[tid + s];
    __syncthreads();
  }
  const float inv = 1.0f / red[0];
#pragma unroll
  for (int j = 0; j < 10; ++j) {
    int l = tid + j * 256;
    malphas[((size_t)b * TT + t) * KL + l] = sv[j] * inv;
  }
}

// ---------------------------------------------------------------------------
// Kernel 5: context[b] (32xD) = malphas[b] (32xKL) @ v[b] (KLxD), f32 WMMA.
// ---------------------------------------------------------------------------
__global__ void context_wmma_kernel(const float* __restrict__ malphas,
                                    const float* __restrict__ vf,
                                    float* __restrict__ ctx) {
  const int lane = threadIdx.x & 31;
  const int wave = threadIdx.x >> 5;
  const int tile = blockIdx.x * 8 + wave;     // 0..127
  const int b  = tile >> 5;
  const int r  = tile & 31;
  const int mt = r >> 4;                      // 0..1
  const int nt = r & 15;                      // 0..15

  const float* A = malphas + (size_t)b * TT * KL;
  const float* B = vf      + (size_t)b * KL * DD;
  const int m0 = mt * 16, n0 = nt * 16;
  const int lm   = lane & 15;
  const int koff = (lane >> 4) << 1;
  const float* arow = A + (size_t)(m0 + lm) * KL + koff;
  const float* bcol = B + (size_t)koff * DD + (n0 + lm);

  v8f acc = {};
  for (int k0 = 0; k0 < KL; k0 += 4) {
    v2f a = *(const v2f*)(arow + k0);
    v2f bm;
    bm.x = bcol[(size_t)k0 * DD];
    bm.y = bcol[(size_t)(k0 + 1) * DD];
    acc = __builtin_amdgcn_wmma_f32_16x16x4_f32(
        false, a, false, bm, (short)0, acc, false, false);
  }

  float* C = ctx + (size_t)b * TT * DD;
  const int col   = n0 + lm;
  const int rbase = m0 + ((lane >> 4) << 3);
#pragma unroll
  for (int rr = 0; rr < 8; ++rr)
    C[(size_t)(rbase + rr) * DD + col] = acc[rr];
}

// ---------------------------------------------------------------------------
// Launch. Inputs: k, v, q, betas, mask, Wq, bq, Wk, we.
// d_out = [context (BS*TT*DD) | modified_alphas (BS*TT*KL)].
// d_ws  = [kproj (BS*KL*DD) | qproj (BS*TT*DD) | scores (BS*KL*TT)] ~12 MB.
// ---------------------------------------------------------------------------
extern "C" void kernel_launch(void* const* d_in, const int* in_sizes, int n_in,
                              void* d_out, int out_size, void* d_ws, size_t ws_size,
                              hipStream_t stream) {
  const float* k_in  = (const float*)d_in[0];
  const float* v_in  = (const float*)d_in[1];
  const float* q_in  = (const float*)d_in[2];
  const float* betas = (const float*)d_in[3];
  const int*   mask  = (const int*)d_in[4];
  const float* Wq    = (const float*)d_in[5];
  const float* bq    = (const float*)d_in[6];
  const float* Wk    = (const float*)d_in[7];
  const float* we    = (const float*)d_in[8];

  float* out     = (float*)d_out;
  float* ctx     = out;                        // BS*TT*DD = 32768
  float* malphas = out + (size_t)BS * TT * DD; // BS*TT*KL = 327680

  float* kproj  = (float*)d_ws;
  float* qproj  = kproj + (size_t)BS * KL * DD;
  float* scores = qproj + (size_t)BS * TT * DD;

  qproj_kernel<<<BS * TT, DD, 0, stream>>>(q_in, Wq, bq, qproj);
  kproj_wmma_kernel<<<(BS * (KL / 16) * (DD / 16)) / 8, 256, 0, stream>>>(k_in, Wk, kproj);
  scores_kernel<<<BS * (KL / 64), 256, 0, stream>>>(kproj, qproj, we, mask, scores);
  softmax_kernel<<<BS * TT, 256, 0, stream>>>(scores, betas, malphas);
  context_wmma_kernel<<<(BS * 2 * 16) / 8, 256, 0, stream>>>(malphas, v_in, ctx);
}